// BinaryClassificationLTC_65712999629564
// MI455X (gfx1250) — compile-verified
//
#include <hip/hip_runtime.h>

// LTC (liquid time-constant) cell, B=8192, S=64, U=128, 6 ODE unfolds.
// Trans-op bound (~872M sigmoid evals); HBM traffic ~20MB (negligible at
// 23.3TB/s). Inner loop reduced to 3 fp ops + 1 trans per synapse pair via
// sigmoid = 0.5*(1+tanh(z/2)) and folding the batch-independent halves
// (sum_i Wdh, sum_i Wnh, sensory equivalents) into per-neuron constants.
// Params are f16x4 packs staged into LDS by the Tensor Data Mover
// (tensor_load_to_lds, confirmed lowered) / async-to-LDS copies; lane=batch
// so every param read is a wave-uniform LDS broadcast.
//
// Workspace layout (bytes):            requirement: ws_size >= ~8.6 MB
//   [0,       131072)  rec params  uint2[16384]   (A,B | Wdh,Wnh) f16 packed
//   [131072,  196608)  sens params uint2[8192]
//   [196608,  198656)  per-neuron  float4[128] (cm_t, num_base, den_base, 0)
//   [198656,  +4MB  )  sensory num float[128*8192]  layout [j][b] (coalesced)
//   [+4MB,    +8MB  )  sensory den float[128*8192]

#define LTC_B 8192
#define LTC_S 64
#define LTC_U 128
#define LTC_UNFOLDS 6

// ---- path selection --------------------------------------------------------
#if defined(__HIP_DEVICE_COMPILE__)
  #if __has_builtin(__builtin_amdgcn_tanhf)
    #define LTC_TANHF __builtin_amdgcn_tanhf
  #elif __has_builtin(__builtin_amdgcn_tanh_f32)
    #define LTC_TANHF __builtin_amdgcn_tanh_f32
  #endif
#endif
#ifdef LTC_TANHF
  #define LTC_USE_TANH 1
#else
  #define LTC_USE_TANH 0
#endif

#if defined(__HIP_DEVICE_COMPILE__) && defined(__gfx1250__)
  #define LTC_GFX1250_DEV 1
#else
  #define LTC_GFX1250_DEV 0
#endif

__device__ __forceinline__ float ltc_rcp(float x) {
#if defined(__HIP_DEVICE_COMPILE__) && __has_builtin(__builtin_amdgcn_rcpf)
  return __builtin_amdgcn_rcpf(x);
#else
  return 1.0f / x;
#endif
}
__device__ __forceinline__ float ltc_exp2(float x) {
#if defined(__HIP_DEVICE_COMPILE__) && __has_builtin(__builtin_amdgcn_exp2f)
  return __builtin_amdgcn_exp2f(x);
#else
  return exp2f(x);
#endif
}

// tanh(t); params always use the tanh formulation: t = 0.5*sigma*(v - mu),
// sigmoid contribution = Wdh*(1 + tanh(t)) with Wdh = 0.5*W.
__device__ __forceinline__ float ltc_th(float t) {
#if LTC_USE_TANH
  return LTC_TANHF(t);                              // 1 trans op
#else
  const float N2LOG2E = -2.8853900817779268f;       // -2*log2(e)
  float e = ltc_exp2(t * N2LOG2E);                  // tanh = 2*sigmoid(2t)-1
  float r = ltc_rcp(1.0f + e);
  return fmaf(2.0f, r, -1.0f);
#endif
}

__device__ __forceinline__ float ltc_softplus(float x) { return logf(1.0f + expf(x)); }

typedef _Float16 ltc_h2 __attribute__((ext_vector_type(2)));
union LtcH2U { unsigned u; ltc_h2 h; };
__device__ __forceinline__ unsigned ltc_pack2(float a, float b) {
  LtcH2U x; x.h = ltc_h2{(_Float16)a, (_Float16)b}; return x.u;
}
__device__ __forceinline__ float ltc_h16(float x) {  // round through f16
  return (float)(_Float16)x;
}

// ---- probe kernel FIRST in file: disasm snippet shows one full pair update
// (reveals v_tanh_f32 vs exp2+rcp, and v_fma_mix_f32 vs separate v_cvt). Never
// launched; exists only for compile-feedback visibility.
__global__ void ltc_probe_sig(const uint2* __restrict__ p_in,
                              const float* __restrict__ v_in,
                              float* __restrict__ o) {
  uint2 p = p_in[threadIdx.x];
  LtcH2U ab; ab.u = p.x; LtcH2U wd; wd.u = p.y;
  float vi = v_in[threadIdx.x];
  float t = fmaf(vi, (float)ab.h.x, (float)ab.h.y);
  float th = ltc_th(t);
  float den = fmaf((float)wd.h.x, th, 1.0f);
  float num = fmaf((float)wd.h.y, th, 2.0f);
  o[threadIdx.x] = num / den;
}

// ---- kernel 3: 6 ODE unfolds + output head --------------------------------
__global__ __launch_bounds__(256) void ltc_main(
    const float* __restrict__ states, const uint2* __restrict__ rec_p_g,
    const float4* __restrict__ jp_g, const float* __restrict__ sn_g,
    const float* __restrict__ sd_g, const float* __restrict__ outw,
    const float* __restrict__ outb, const float* __restrict__ linW,
    const float* __restrict__ linb, float* __restrict__ out) {
  __shared__ uint2 rp[LTC_U * LTC_U];   // 128 KB fused recurrent params
  __shared__ float vs[LTC_U * 32];      //  16 KB state exchange [i][lane]
  __shared__ float4 jp[LTC_U];          //   2 KB
  const int tid = threadIdx.x, lane = tid & 31, wave = tid >> 5;
  const int b = blockIdx.x * 32 + lane;

#if LTC_GFX1250_DEV
  // TDM stages the full 128KB param block: 1D tile, 16384 x 8B elements.
  // (Confirmed lowered: tensor_load_to_lds s[g0], s[g1].)
  if (wave == 0) {
    typedef unsigned v4u __attribute__((ext_vector_type(4)));
    typedef int v4i __attribute__((ext_vector_type(4)));
    typedef int v8i __attribute__((ext_vector_type(8)));
    unsigned lds_addr = (unsigned)(unsigned long long)(void*)&rp[0];
    unsigned long long ga = (unsigned long long)(const void*)rec_p_g;
    v4u g0 = {1u,                                   // count=1
              lds_addr,                             // lds_addr [63:32]
              (unsigned)(ga & 0xFFFFFFFFull),       // global_addr lo
              (unsigned)((ga >> 32) & 0x1FFFFFFull) | (2u << 30)}; // type=2
    v8i g1 = {(int)(3u << 16),          // data_size = 8B, no mask/pad/iter
              (int)(16384u << 16),      // tensor_dim0 lo16 = 16384
              (int)(1u << 16),          // tensor_dim0 hi=0; tensor_dim1 lo16=1
              (int)(16384u << 16),      // tensor_dim1 hi=0; tile_dim0 = 16384
              0,                        // tile_dim1/2 = 0 (unused)
              16384,                    // tensor_dim0_stride lo32
              0, 0};
    v4i gz = {0, 0, 0, 0};
#if __clang_major__ >= 23
    v8i gz8 = {0, 0, 0, 0, 0, 0, 0, 0};
    __builtin_amdgcn_tensor_load_to_lds(g0, g1, gz, gz, gz8, 0);
#else
    __builtin_amdgcn_tensor_load_to_lds(g0, g1, gz, gz, 0);
#endif
    __builtin_amdgcn_s_wait_tensorcnt(0);
  }
#else
  for (int e = tid; e < LTC_U * LTC_U; e += 256) rp[e] = rec_p_g[e];
#endif

  __builtin_prefetch(&sn_g[wave * 16 * LTC_B + b], 0, 1);

  if (tid < LTC_U) jp[tid] = jp_g[tid];
  for (int i = wave; i < LTC_U; i += 8) vs[i * 32 + lane] = states[b * LTC_U + i];

  float sn[16], sd[16];
#pragma unroll
  for (int jj = 0; jj < 16; jj++) {
    int j = wave * 16 + jj;
    sn[jj] = sn_g[j * LTC_B + b];
    sd[jj] = sd_g[j * LTC_B + b];
  }
  __syncthreads();

  float vnew[16];
  for (int step = 0; step < LTC_UNFOLDS; step++) {
    float num[16], den[16];
#pragma unroll
    for (int jj = 0; jj < 16; jj++) { num[jj] = sn[jj]; den[jj] = sd[jj]; }

    for (int i = 0; i < LTC_U; i++) {
      float vi = vs[i * 32 + lane];                 // conflict-free per-lane
#pragma unroll
      for (int jj = 0; jj < 16; jj++) {
        uint2 p = rp[i * LTC_U + wave * 16 + jj];   // wave-uniform broadcast
        LtcH2U ab; ab.u = p.x; LtcH2U wd; wd.u = p.y;
        float t = fmaf(vi, (float)ab.h.x, (float)ab.h.y);  // fma_mix
        float th = ltc_th(t);                              // v_tanh_f32
        den[jj] = fmaf((float)wd.h.x, th, den[jj]);        // fma_mix
        num[jj] = fmaf((float)wd.h.y, th, num[jj]);        // fma_mix
      }
    }
#pragma unroll
    for (int jj = 0; jj < 16; jj++) {
      int j = wave * 16 + jj;
      float4 q = jp[j];                 // q.y/q.z hold all baseline terms
      float vj = vs[j * 32 + lane];
      vnew[jj] = (q.x * vj + q.y + num[jj]) / (q.z + den[jj]);
    }
    __syncthreads();
#pragma unroll
    for (int jj = 0; jj < 16; jj++) vs[(wave * 16 + jj) * 32 + lane] = vnew[jj];
    __syncthreads();
  }

  float* out_v = out + LTC_B;                       // tuple: (probs[B], v[B,U])
#pragma unroll
  for (int jj = 0; jj < 16; jj++) out_v[b * LTC_U + wave * 16 + jj] = vnew[jj];
  if (wave == 0) {                                  // j==0 is the motor neuron
    float motor = fmaf(vnew[0], outw[0], outb[0]);
    float logit = fmaf(motor, linW[0], linb[0]);
    out[b] = 1.0f / (1.0f + expf(-logit));
  }
}

// ---- kernel 2: sensory th-weighted sums -> ws ([j][b], coalesced) ----------
__global__ __launch_bounds__(256) void ltc_sensory(
    const float* __restrict__ inputs, const float* __restrict__ iw,
    const float* __restrict__ ib, const uint2* __restrict__ sens_p,
    float* __restrict__ out_num, float* __restrict__ out_den) {
  __shared__ uint2 sp_lds[LTC_S * LTC_U];   // 64 KB
  __shared__ float x_lds[LTC_S * 32];       //  8 KB
  const int tid = threadIdx.x, lane = tid & 31, wave = tid >> 5;
  const int b0 = blockIdx.x * 32;

#if LTC_GFX1250_DEV
  // gfx1250 async-to-LDS staging (ASYNCcnt path, no VGPR round-trip)
  for (int e = tid; e < LTC_S * LTC_U; e += 256) {
    unsigned lds_a = (unsigned)(unsigned long long)(void*)&sp_lds[e];
    const uint2* gp = &sens_p[e];
    asm volatile("global_load_async_to_lds_b64 %0, %1, off"
                 :: "v"(lds_a), "v"(gp) : "memory");
  }
  asm volatile("s_wait_asynccnt 0" ::: "memory");
#else
  for (int e = tid; e < LTC_S * LTC_U; e += 256) sp_lds[e] = sens_p[e];
#endif
  for (int e = tid; e < LTC_S * 32; e += 256) {
    int s = e >> 5, l = e & 31;
    x_lds[e] = fmaf(inputs[(b0 + l) * LTC_S + s], iw[s], ib[s]);
  }
  __syncthreads();

  float num[16], den[16];
#pragma unroll
  for (int jj = 0; jj < 16; jj++) { num[jj] = 0.0f; den[jj] = 0.0f; }

  for (int s = 0; s < LTC_S; s++) {
    float xs = x_lds[s * 32 + lane];
#pragma unroll
    for (int jj = 0; jj < 16; jj++) {
      uint2 p = sp_lds[s * LTC_U + wave * 16 + jj];   // wave-uniform broadcast
      LtcH2U ab; ab.u = p.x; LtcH2U wd; wd.u = p.y;
      float t = fmaf(xs, (float)ab.h.x, (float)ab.h.y);
      float th = ltc_th(t);
      den[jj] = fmaf((float)wd.h.x, th, den[jj]);
      num[jj] = fmaf((float)wd.h.y, th, num[jj]);
    }
  }
  const int b = b0 + lane;
#pragma unroll
  for (int jj = 0; jj < 16; jj++) {
    int j = wave * 16 + jj;
    out_num[j * LTC_B + b] = num[jj];
    out_den[j * LTC_B + b] = den[jj];
  }
}

// ---- kernel 1: fuse + pack parameters, fold baselines ----------------------
__global__ __launch_bounds__(256) void ltc_prep(
    const float* __restrict__ smu, const float* __restrict__ ssig,
    const float* __restrict__ sw,  const float* __restrict__ serev,
    const float* __restrict__ smask,
    const float* __restrict__ gleak, const float* __restrict__ vleak,
    const float* __restrict__ cm,
    const float* __restrict__ mu, const float* __restrict__ sig,
    const float* __restrict__ w,  const float* __restrict__ erev,
    const float* __restrict__ mask,
    uint2* __restrict__ rec_p, uint2* __restrict__ sens_p,
    float4* __restrict__ jp) {
  int t = blockIdx.x * blockDim.x + threadIdx.x;
  if (t < LTC_U * LTC_U) {
    float A = 0.5f * sig[t];                    // t = 0.5*sigma*(v - mu)
    float Bc = -A * mu[t];
    float Wdh = 0.5f * ltc_softplus(w[t]) * mask[t];
    float Wnh = Wdh * erev[t];
    rec_p[t] = make_uint2(ltc_pack2(A, Bc), ltc_pack2(Wdh, Wnh));
  } else if (t < LTC_U * LTC_U + LTC_S * LTC_U) {
    int e = t - LTC_U * LTC_U;
    float A = 0.5f * ssig[e];
    float Bc = -A * smu[e];
    float Wdh = 0.5f * ltc_softplus(sw[e]) * smask[e];
    float Wnh = Wdh * serev[e];
    sens_p[e] = make_uint2(ltc_pack2(A, Bc), ltc_pack2(Wdh, Wnh));
  } else if (t < LTC_U * LTC_U + LTC_S * LTC_U + LTC_U) {
    int j = t - (LTC_U * LTC_U + LTC_S * LTC_U);
    float cmt = ltc_softplus(cm[j]) * (float)LTC_UNFOLDS;
    float gp  = ltc_softplus(gleak[j]);
    // batch-independent halves of the synapse sums (through f16 rounding so
    // they match the packed weights exactly)
    float bn = 0.0f, bd = 0.0f;
    for (int i = 0; i < LTC_U; i++) {
      float h = ltc_h16(0.5f * ltc_softplus(w[i * LTC_U + j]) * mask[i * LTC_U + j]);
      bd += h;
      bn += h * erev[i * LTC_U + j];
    }
    for (int s = 0; s < LTC_S; s++) {
      float h = ltc_h16(0.5f * ltc_softplus(sw[s * LTC_U + j]) * smask[s * LTC_U + j]);
      bd += h;
      bn += h * serev[s * LTC_U + j];
    }
    jp[j] = make_float4(cmt, gp * vleak[j] + bn, cmt + gp + 1e-8f + bd, 0.0f);
  }
}

// ---- launcher --------------------------------------------------------------
extern "C" void kernel_launch(void* const* d_in, const int* in_sizes, int n_in,
                              void* d_out, int out_size, void* d_ws,
                              size_t ws_size, hipStream_t stream) {
  (void)in_sizes; (void)n_in; (void)out_size; (void)ws_size;
  const float* inputs  = (const float*)d_in[0];
  const float* states  = (const float*)d_in[1];
  const float* input_w = (const float*)d_in[2];
  const float* input_b = (const float*)d_in[3];
  const float* smu     = (const float*)d_in[4];
  const float* ssig    = (const float*)d_in[5];
  const float* sw      = (const float*)d_in[6];
  const float* serev   = (const float*)d_in[7];
  const float* smask   = (const float*)d_in[8];
  const float* gleak   = (const float*)d_in[9];
  const float* vleak   = (const float*)d_in[10];
  const float* cm      = (const float*)d_in[11];
  const float* mu      = (const float*)d_in[12];
  const float* sig     = (const float*)d_in[13];
  const float* w       = (const float*)d_in[14];
  const float* erev    = (const float*)d_in[15];
  const float* mask    = (const float*)d_in[16];
  const float* outw    = (const float*)d_in[17];
  const float* outb    = (const float*)d_in[18];
  const float* linW    = (const float*)d_in[19];
  const float* linb    = (const float*)d_in[20];
  float* out = (float*)d_out;

  char* ws = (char*)d_ws;
  uint2*  rec_p  = (uint2*)(ws);
  uint2*  sens_p = (uint2*)(ws + 131072);
  float4* jp     = (float4*)(ws + 196608);
  float*  sn     = (float*)(ws + 198656);
  float*  sd     = (float*)(ws + 198656 + 4194304);

  const int prep_n = LTC_U * LTC_U + LTC_S * LTC_U + LTC_U;
  ltc_prep<<<(prep_n + 255) / 256, 256, 0, stream>>>(
      smu, ssig, sw, serev, smask, gleak, vleak, cm, mu, sig, w, erev, mask,
      rec_p, sens_p, jp);
  ltc_sensory<<<LTC_B / 32, 256, 0, stream>>>(inputs, input_w, input_b, sens_p,
                                              sn, sd);
  ltc_main<<<LTC_B / 32, 256, 0, stream>>>(states, rec_p, jp, sn, sd, outw,
                                           outb, linW, linb, out);
}